// LowPassRadio_28870770163739
// MI455X (gfx1250) — compile-verified
//
#include <hip/hip_runtime.h>
#include <math.h>

// -------- problem constants --------
constexpr int TLEN  = 320000;   // samples per row
constexpr int BROWS = 64;       // batch rows
constexpr int CHUNK = 320;      // samples per chunk (multiple of 16)
constexpr int NC    = TLEN / CHUNK;  // 1000 chunks per row
constexpr int NBLK  = CHUNK / 16;    // 20 WMMA blocks per chunk

typedef __attribute__((ext_vector_type(2))) float v2f;
typedef __attribute__((ext_vector_type(8))) float v8f;

// FIR front-end: f(t) = b0*x[t] + b1*x[t-1] + b2*x[t-2], zero history before t=0
__device__ __forceinline__ float fir3(const float* __restrict__ xr, int t,
                                      float b0, float b1, float b2) {
    float x0 = xr[t];
    float x1 = (t >= 1) ? xr[t - 1] : 0.0f;
    float x2 = (t >= 2) ? xr[t - 2] : 0.0f;
    return fmaf(b0, x0, fmaf(b1, x1, b2 * x2));
}

// -------- Phase 1: per-chunk terminal state with zero entry state --------
__global__ void lp_phase1(const float* __restrict__ x, float2* __restrict__ dstate,
                          float b0, float b1, float b2, float a1, float a2) {
    int gid = blockIdx.x * blockDim.x + threadIdx.x;
    if (gid >= BROWS * NC) return;
    int row = gid / NC;
    int c   = gid % NC;
    const float* xr = x + (long)row * TLEN;
    int t0 = c * CHUNK;
    float y1 = 0.0f, y2 = 0.0f;
    float xm1 = (t0 >= 1) ? xr[t0 - 1] : 0.0f;
    float xm2 = (t0 >= 2) ? xr[t0 - 2] : 0.0f;
    for (int n = 0; n < CHUNK; ++n) {
        float xn = xr[t0 + n];
        float f  = fmaf(b0, xn, fmaf(b1, xm1, b2 * xm2));
        float y  = fmaf(-a1, y1, fmaf(-a2, y2, f));
        y2 = y1; y1 = y; xm2 = xm1; xm1 = xn;
    }
    dstate[gid] = make_float2(y1, y2);   // (y at chunk-end, y at chunk-end-1)
}

// -------- Phase 2: serial combine over chunks (2x2 affine scan), one thread/row --------
__global__ void lp_phase2(const float2* __restrict__ dstate, float2* __restrict__ entry,
                          float p00, float p01, float p10, float p11) {
    int row = blockIdx.x * blockDim.x + threadIdx.x;
    if (row >= BROWS) return;
    float s1 = 0.0f, s2 = 0.0f;   // (y[t0-1], y[t0-2]) entering next chunk
    for (int c = 0; c < NC; ++c) {
        entry[row * NC + c] = make_float2(s1, s2);
        float2 d = dstate[row * NC + c];
        float n1 = fmaf(p00, s1, fmaf(p01, s2, d.x));
        float n2 = fmaf(p10, s1, fmaf(p11, s2, d.y));
        s1 = n1; s2 = n2;
    }
}

// -------- Phase 3: WMMA block reconstruction --------
// One wave handles 16 rows x one chunk. Per 16-step block:
//   Y(16x16) = Aaug(16x20) x Baug(20x16)  via 5x V_WMMA_F32_16X16X4_F32
//   A: M=row, K<16 -> f(tb+K);  K=16,17 -> entry state (ym1, ym2); K=18,19 -> 0
//   B: N=time, K<16 -> h[t-K] (impulse response, lower-tri); K=16 -> g1[t]=h[t+1];
//      K=17 -> g2[t]=-a2*h[t]; K=18,19 -> 0
__global__ __launch_bounds__(256) void lp_phase3(
    const float* __restrict__ x, float* __restrict__ out,
    const float2* __restrict__ entry,
    float b0, float b1, float b2, float a1, float a2) {

    int lane = threadIdx.x & 31;
    int wid  = blockIdx.x * 8 + (threadIdx.x >> 5);
    int c  = wid % NC;          // chunk index
    int rg = wid / NC;          // row group 0..3
    if (rg >= BROWS / 16) return;   // wave-uniform guard (EXEC stays all-ones)

    int half = lane >> 4;
    int lid  = lane & 15;
    int r0   = rg * 16;
    int row  = r0 + lid;        // A-matrix M index; also B's time index t = lid
    int t0   = c * CHUNK;

    // Impulse response h[0..16] of y_n = -a1 y_{n-1} - a2 y_{n-2} + delta_n
    float h[17];
    h[0] = 1.0f;
    h[1] = -a1;
    #pragma unroll
    for (int n = 2; n <= 16; ++n) h[n] = fmaf(-a1, h[n - 1], -a2 * h[n - 2]);

    // Column values of augmented T for this lane's time row t = lid
    float Tt[20];
    #pragma unroll
    for (int j = 0; j < 16; ++j) Tt[j] = (lid >= j) ? h[lid - j] : 0.0f;
    Tt[16] = h[lid + 1];        // g1[t]
    Tt[17] = -a2 * h[lid];      // g2[t]
    Tt[18] = 0.0f;
    Tt[19] = 0.0f;

    // B operand registers (constant across blocks).
    // B VGPR layout (4x16 f32): vgpr0 lanes0-15 K=4kc+0, lanes16-31 K=4kc+2;
    //                           vgpr1 lanes0-15 K=4kc+1, lanes16-31 K=4kc+3.
    v2f Bk[5];
    #pragma unroll
    for (int kc = 0; kc < 5; ++kc) {
        Bk[kc].x = half ? Tt[4 * kc + 2] : Tt[4 * kc + 0];
        Bk[kc].y = half ? Tt[4 * kc + 3] : Tt[4 * kc + 1];
    }

    float2 e = entry[row * NC + c];
    float ym1 = e.x, ym2 = e.y;

    const float* xr = x + (long)row * TLEN;

    for (int blk = 0; blk < NBLK; ++blk) {
        int tb = t0 + blk * 16;

        // Build A operand (16x4 f32 per step: vgpr0 K=4kc+{0|2}, vgpr1 K=4kc+{1|3})
        v2f Ak[5];
        #pragma unroll
        for (int kc = 0; kc < 4; ++kc) {
            int k0 = 4 * kc + (half ? 2 : 0);
            int k1 = 4 * kc + (half ? 3 : 1);
            Ak[kc].x = fir3(xr, tb + k0, b0, b1, b2);
            Ak[kc].y = fir3(xr, tb + k1, b0, b1, b2);
        }
        Ak[4].x = half ? 0.0f : ym1;   // K=16 (states live in lanes 0-15)
        Ak[4].y = half ? 0.0f : ym2;   // K=17

        v8f acc = {};
        #pragma unroll
        for (int kc = 0; kc < 5; ++kc) {
            acc = __builtin_amdgcn_wmma_f32_16x16x4_f32(
                /*neg_a=*/false, Ak[kc], /*neg_b=*/false, Bk[kc],
                /*c_mod=*/(short)0, acc, /*reuse_a=*/false, /*reuse_b=*/false);
        }

        // Store: D vgpr v holds Y[M=v+8*half][N=lid] -> coalesced 64B runs per half-wave
        #pragma unroll
        for (int v = 0; v < 8; ++v) {
            out[(long)(r0 + v + 8 * half) * TLEN + tb + lid] = acc[v];
        }

        // Extract new entry state for next block: ym1 = Y[row][15], ym2 = Y[row][14]
        float nym1 = 0.0f, nym2 = 0.0f;
        #pragma unroll
        for (int v = 0; v < 8; ++v) {
            float lo15 = __shfl(acc[v], 15, 32);
            float hi15 = __shfl(acc[v], 31, 32);
            float lo14 = __shfl(acc[v], 14, 32);
            float hi14 = __shfl(acc[v], 30, 32);
            if (lid == v)     { nym1 = lo15; nym2 = lo14; }
            if (lid == v + 8) { nym1 = hi15; nym2 = hi14; }
        }
        ym1 = nym1; ym2 = nym2;
    }
}

extern "C" void kernel_launch(void* const* d_in, const int* in_sizes, int n_in,
                              void* d_out, int out_size, void* d_ws, size_t ws_size,
                              hipStream_t stream) {
    const float* x = (const float*)d_in[0];
    float* out = (float*)d_out;

    // Workspace carve-up: dstate then entry, each BROWS*NC float2
    float2* dstate = (float2*)d_ws;
    float2* entry  = dstate + (size_t)BROWS * NC;

    // Host-side coefficient computation (double precision)
    double w0    = 2.0 * M_PI * 7500.0 / 16000.0;
    double alpha = sin(w0) / (2.0 * 0.707);
    double cw    = cos(w0);
    double b0 = (1.0 - cw) / 2.0, b1 = 1.0 - cw, b2 = b0;
    double a0 = 1.0 + alpha, a1 = -2.0 * cw, a2 = 1.0 - alpha;
    b0 /= a0; b1 /= a0; b2 /= a0; a1 /= a0; a2 /= a0;

    // P = A^CHUNK where A = [[-a1,-a2],[1,0]] (state = [y_n, y_{n-1}])
    double p00 = 1, p01 = 0, p10 = 0, p11 = 1;
    for (int i = 0; i < CHUNK; ++i) {
        double n00 = -a1 * p00 - a2 * p10;
        double n01 = -a1 * p01 - a2 * p11;
        double n10 = p00, n11 = p01;
        p00 = n00; p01 = n01; p10 = n10; p11 = n11;
    }

    float fb0 = (float)b0, fb1 = (float)b1, fb2 = (float)b2;
    float fa1 = (float)a1, fa2 = (float)a2;

    int n1 = BROWS * NC;
    lp_phase1<<<(n1 + 255) / 256, 256, 0, stream>>>(x, dstate, fb0, fb1, fb2, fa1, fa2);
    lp_phase2<<<1, 64, 0, stream>>>(dstate, entry,
                                    (float)p00, (float)p01, (float)p10, (float)p11);
    int nwaves = NC * (BROWS / 16);          // 4000 waves
    lp_phase3<<<nwaves / 8, 256, 0, stream>>>(x, out, entry, fb0, fb1, fb2, fa1, fa2);
}